// ChannelAttention_7361573945544
// MI455X (gfx1250) — compile-verified
//
#include <hip/hip_runtime.h>
#include <float.h>
#include <math.h>

typedef __attribute__((ext_vector_type(2))) float v2f;
typedef __attribute__((ext_vector_type(4))) float v4f;
typedef __attribute__((ext_vector_type(8))) float v8f;

#define B_ 8
#define S_ 4096
#define C_ 1024
#define H_ 64
#define NCHUNK 64
#define ROWS (S_ / NCHUNK)          // 64 rows per chunk

// workspace layout (in floats)
#define WS_PSUM 0                                  // [B][NCHUNK][C]
#define WS_PMAX (B_ * NCHUNK * C_)                 // [B][NCHUNK][C]
#define WS_PCNT (2 * B_ * NCHUNK * C_)             // [B][NCHUNK]
#define WS_V    (WS_PCNT + B_ * NCHUNK)            // [16][C]  rows 0-7 mean, 8-15 max
#define WS_A    (WS_V + 16 * C_)                   // [8][C]   sigmoid gate

__device__ inline v4f max4(v4f a, v4f b) {
    v4f r;
    r.x = fmaxf(a.x, b.x); r.y = fmaxf(a.y, b.y);
    r.z = fmaxf(a.z, b.z); r.w = fmaxf(a.w, b.w);
    return r;
}

// ---------------------------------------------------------------------------
// Kernel 1: per-(batch, S-chunk) masked partial sum / max over 64 rows.
// Mask is uniform per row -> uniform branch skips the VMEM load entirely
// for unmasked rows (halves phase-1 HBM traffic on average).
// ---------------------------------------------------------------------------
__global__ __launch_bounds__(256) void k_partial(const float* __restrict__ x,
                                                 const int* __restrict__ mask,
                                                 float* __restrict__ ws) {
    __shared__ int smask[ROWS];
    const int b  = blockIdx.x / NCHUNK;
    const int ch = blockIdx.x % NCHUNK;
    const int t  = threadIdx.x;

    if (t < ROWS) smask[t] = mask[b * S_ + ch * ROWS + t];
    __syncthreads();

    const float* xb = x + ((size_t)b * S_ + (size_t)ch * ROWS) * C_ + 4 * t;

    v4f sum = (v4f)(0.0f);
    v4f mx  = (v4f)(-FLT_MAX);
    int cnt = 0;

#pragma unroll 4
    for (int s = 0; s < ROWS; ++s) {
        if (smask[s]) {                       // uniform across the block
            v4f v = *(const v4f*)(xb + (size_t)s * C_);
            sum += v;
            mx = max4(mx, v);
            ++cnt;
        }
    }

    const size_t po = ((size_t)b * NCHUNK + ch) * C_ + 4 * t;
    *(v4f*)(ws + WS_PSUM + po) = sum;
    *(v4f*)(ws + WS_PMAX + po) = mx;
    if (t == 0) ws[WS_PCNT + b * NCHUNK + ch] = (float)cnt;
}

// ---------------------------------------------------------------------------
// Kernel 2: fold the 64 partials per (b,c); build V[16][1024] (mean on top,
// max below) for the stacked WMMA MLP.
// ---------------------------------------------------------------------------
__global__ __launch_bounds__(256) void k_combine(float* __restrict__ ws) {
    const int idx = blockIdx.x * 256 + threadIdx.x;   // 0 .. B*C-1
    const int b = idx >> 10;
    const int c = idx & (C_ - 1);

    float s = 0.0f, mx = -FLT_MAX, cntf = 0.0f;
#pragma unroll 4
    for (int ch = 0; ch < NCHUNK; ++ch) {
        const size_t po = ((size_t)b * NCHUNK + ch) * C_ + c;
        s    += ws[WS_PSUM + po];
        mx    = fmaxf(mx, ws[WS_PMAX + po]);
        cntf += ws[WS_PCNT + b * NCHUNK + ch];
    }
    const float mean = s / fmaxf(cntf, 1.0f);
    ws[WS_V + b * C_ + c]        = mean;   // rows 0-7: masked mean
    ws[WS_V + (b + 8) * C_ + c]  = mx;     // rows 8-15: masked max (-FLT_MAX == finfo.min)
}

// ---------------------------------------------------------------------------
// Kernel 3: stacked MLP with V_WMMA_F32_16X16X4_F32 (exact fp32).
//   H  = relu(V @ W0^T)        : M=16, N=64,  K=1024  (waves 0-3, one N-tile each)
//   O  = H @ W1^T              : M=16, N=1024, K=64   (8 waves, 8 N-tiles each)
//   a[b][c] = sigmoid(O[b][c] + O[b+8][c])  via shfl_xor(16) on the D fragment
// Fragment layouts per CDNA5 ISA 7.12.2:
//   A (16x4 f32):  lane -> M = lane&15, VGPR j -> K = 2*(lane>>4)+j
//   B (4x16 f32):  lane -> N = lane&15, VGPR j -> K = 2*(lane>>4)+j
//   C/D (16x16):   lane,VGPR r -> M = r + 8*(lane>>4), N = lane&15
// ---------------------------------------------------------------------------
#define HSTRIDE 68   // 16x64 H tile in LDS, padded stride -> conflict-free rows

__global__ __launch_bounds__(256) void k_mlp(const float* __restrict__ W0,
                                             const float* __restrict__ W1,
                                             float* __restrict__ ws) {
    __shared__ float Hm[16 * HSTRIDE];

    const int tid  = threadIdx.x;
    const int wave = tid >> 5;
    const int lane = tid & 31;
    const int m    = lane & 15;
    const int koff = (lane >> 4) << 1;     // 0 or 2
    const float* V = ws + WS_V;

    // ---- GEMM1: H = relu(V @ W0^T), 4 N-tiles on waves 0..3 ----
    if (wave < 4) {
        const int n = wave * 16 + m;
        const float* arow = V  + (size_t)m * C_;
        const float* brow = W0 + (size_t)n * C_;
        v8f acc = (v8f)(0.0f);
        for (int kk = 0; kk < C_ / 4; ++kk) {
            const int k0 = kk * 4 + koff;
            v2f af = *(const v2f*)(arow + k0);
            v2f bf = *(const v2f*)(brow + k0);
            acc = __builtin_amdgcn_wmma_f32_16x16x4_f32(
                false, af, false, bf, (short)0, acc, false, false);
        }
        const int rbase = (lane >> 4) * 8;
#pragma unroll
        for (int r = 0; r < 8; ++r)
            Hm[(rbase + r) * HSTRIDE + n] = fmaxf(acc[r], 0.0f);
    }
    __syncthreads();

    // ---- GEMM2: O = H @ W1^T, 64 N-tiles over 8 waves ----
#pragma unroll 1
    for (int tt = 0; tt < 8; ++tt) {
        const int n = (wave * 8 + tt) * 16 + m;
        const float* brow = W1 + (size_t)n * H_;
        v8f acc = (v8f)(0.0f);
#pragma unroll
        for (int kk = 0; kk < H_ / 4; ++kk) {
            const int k0 = kk * 4 + koff;
            v2f af = *(const v2f*)(&Hm[m * HSTRIDE + k0]);
            v2f bf = *(const v2f*)(brow + k0);
            acc = __builtin_amdgcn_wmma_f32_16x16x4_f32(
                false, af, false, bf, (short)0, acc, false, false);
        }
#pragma unroll
        for (int r = 0; r < 8; ++r) {
            const float v = acc[r];
            const float ssum = v + __shfl_xor(v, 16, 32);  // O[r][n] + O[r+8][n]
            if (lane < 16)
                ws[WS_A + r * C_ + n] = 1.0f / (1.0f + expf(-ssum));
        }
    }
}

// ---------------------------------------------------------------------------
// Kernel 4: out[b,s,c] = x[b,s,c] * a[b,c].
// x read temporal (stays resident in 192MB L2 across passes/replays);
// out stored non-temporal (write-once, do not evict x).
// ---------------------------------------------------------------------------
__global__ __launch_bounds__(256) void k_scale(const float* __restrict__ x,
                                               const float* __restrict__ ws,
                                               float* __restrict__ out) {
    const int t = threadIdx.x;
    const int blocksPerB = S_ / 4;                 // 4 rows per block
    const int b    = blockIdx.x / blocksPerB;
    const int row0 = (blockIdx.x % blocksPerB) * 4;

    const v4f av = *(const v4f*)(ws + WS_A + b * C_ + 4 * t);
    const size_t base = ((size_t)b * S_ + row0) * C_ + 4 * t;

#pragma unroll
    for (int r = 0; r < 4; ++r) {
        v4f v = *(const v4f*)(x + base + (size_t)r * C_);
        v *= av;
        __builtin_nontemporal_store(v, (v4f*)(out + base + (size_t)r * C_));
    }
}

// ---------------------------------------------------------------------------
extern "C" void kernel_launch(void* const* d_in, const int* in_sizes, int n_in,
                              void* d_out, int out_size, void* d_ws, size_t ws_size,
                              hipStream_t stream) {
    const float* x    = (const float*)d_in[0];
    const int*   mask = (const int*)d_in[1];
    const float* W0   = (const float*)d_in[2];
    const float* W1   = (const float*)d_in[3];
    float* out = (float*)d_out;
    float* ws  = (float*)d_ws;

    k_partial<<<B_ * NCHUNK, 256, 0, stream>>>(x, mask, ws);
    k_combine<<<(B_ * C_) / 256, 256, 0, stream>>>(ws);
    k_mlp<<<1, 256, 0, stream>>>(W0, W1, ws);
    k_scale<<<B_ * (S_ / 4), 256, 0, stream>>>(x, ws, out);
}